// FastVoxelGen_46162308497693
// MI455X (gfx1250) — compile-verified
//
#include <hip/hip_runtime.h>
#include <hip/hip_bf16.h>

// FastVoxelGen: 6-camera -> 640K-query feature gather.
// Memory-bound (no matmul structure -> WMMA not applicable).
// Strategy: coalesced 16B NT streaming stores (328 MB write-once),
// scattered gather reads stay L2-resident (img_feats = 34.6 MB << 192 MB L2),
// gfx1250 async-to-LDS staging of the validity mask + global prefetch.

typedef __attribute__((ext_vector_type(4))) float v4f;  // native vector for NT store

namespace {
constexpr int kNQ  = 200 * 200 * 16;  // 640000 queries
constexpr int kCam = 6;
constexpr int kC   = 128;
constexpr int kH   = 64;
constexpr int kW   = 176;
constexpr int kHW  = kH * kW;         // 11264 (channel stride in floats)
constexpr int kCHW = kC * kHW;        // 1441792 (camera stride in floats)
constexpr int kQPT = 4;               // queries per thread (16B stores)
constexpr int kTPB = 256;             // 8 wave32 per block
constexpr int kQPB = kQPT * kTPB;     // 1024 queries per block
}

__global__ __launch_bounds__(kTPB)
void voxel_gather_kernel(const float* __restrict__ img,     // [6][128][64][176]
                         const float* __restrict__ points,  // [6][640000][2]
                         const int*   __restrict__ valid,   // [6][640000]
                         float* __restrict__ out)           // [128][640000]
{
  __shared__ int sValid[kCam * kQPB];  // 24 KB of the 320 KB WGP LDS

  const int tid = threadIdx.x;
  const int q0  = blockIdx.x * kQPB + tid * kQPT;

  // --- Stage this block's validity rows into LDS with gfx1250 async copies.
  // Per-lane: copies 16 bytes from the global address in vaddr to the LDS address in vdst (4 queries' flags).
  // Tracked by ASYNCcnt; waited with s_wait_asynccnt.
  #pragma unroll
  for (int cam = 0; cam < kCam; ++cam) {
    const int* gsrc = valid + (size_t)cam * kNQ + q0;
    unsigned lds_addr = (unsigned)(size_t)&sValid[cam * kQPB + tid * kQPT];
    unsigned long long gaddr = (unsigned long long)(size_t)gsrc;
    asm volatile("global_load_async_to_lds_b128 %0, %1, off"
                 :: "v"(lds_addr), "v"(gaddr)
                 : "memory");
    // Warm L2 for the strided points gather below (global_prefetch_b8).
    __builtin_prefetch(points + ((size_t)cam * kNQ + q0) * 2, 0, 0);
  }
  asm volatile("s_wait_asynccnt 0" ::: "memory");
  __syncthreads();

  // --- Per-query camera selection (reference: max cam index with valid set).
  int   base[kQPT];
  float msk[kQPT];
  #pragma unroll
  for (int j = 0; j < kQPT; ++j) {
    const int q = q0 + j;
    int sel = -1;
    #pragma unroll
    for (int cam = 0; cam < kCam; ++cam)
      if (sValid[cam * kQPB + tid * kQPT + j] != 0) sel = cam;  // keeps max
    msk[j] = (sel >= 0) ? 1.0f : 0.0f;
    const int cam = (sel >= 0) ? sel : 0;  // clamp -> address always in-range
    const float px = points[((size_t)cam * kNQ + q) * 2 + 0];
    const float py = points[((size_t)cam * kNQ + q) * 2 + 1];
    const int x = __float2int_rn(px);   // RNE, matches jnp.round
    const int y = __float2int_rn(py);
    base[j] = cam * kCHW + y * kW + x;  // img[cam][c][y][x] = base + c*kHW
  }

  const float* p0 = img + base[0];
  const float* p1 = img + base[1];
  const float* p2 = img + base[2];
  const float* p3 = img + base[3];
  float* op = out + q0;

  // --- Channel loop: 4 scattered L2-resident dword loads -> one coalesced
  // 16B non-temporal streaming store per thread per channel.
  // Unroll 8: channel offsets (c*45056 B) fit the signed 24-bit immediate,
  // and ~32 loads stay in flight per thread.
  #pragma unroll 8
  for (int c = 0; c < kC; ++c) {
    v4f v;
    v.x = p0[(size_t)c * kHW] * msk[0];
    v.y = p1[(size_t)c * kHW] * msk[1];
    v.z = p2[(size_t)c * kHW] * msk[2];
    v.w = p3[(size_t)c * kHW] * msk[3];
    __builtin_nontemporal_store(v, reinterpret_cast<v4f*>(op + (size_t)c * kNQ));
  }
}

extern "C" void kernel_launch(void* const* d_in, const int* in_sizes, int n_in,
                              void* d_out, int out_size, void* d_ws, size_t ws_size,
                              hipStream_t stream) {
  (void)in_sizes; (void)n_in; (void)out_size; (void)d_ws; (void)ws_size;
  const float* img    = (const float*)d_in[0];  // img_feats, float32
  const float* points = (const float*)d_in[1];  // points, float32
  const int*   valid  = (const int*)d_in[2];    // bool -> int32 per harness
  float* out = (float*)d_out;                   // [128][200][200][16] float32

  dim3 grid(kNQ / kQPB);   // 640000 / 1024 = 625 blocks, 8 waves each
  dim3 block(kTPB);
  voxel_gather_kernel<<<grid, block, 0, stream>>>(img, points, valid, out);
}